// get_model_20134806684320
// MI455X (gfx1250) — compile-verified
//
#include <hip/hip_runtime.h>
#include <hip/hip_bf16.h>

typedef __attribute__((ext_vector_type(2))) float v2f;
typedef __attribute__((ext_vector_type(8))) float v8f;

#define NPTS 16384
#define BATCH 64

// ---------------------------------------------------------------------------
// Wave-level fp32 GEMM layer using V_WMMA_F32_16X16X4_F32.
//   Hout(MxN) = relu( Hin(MxKpad) * W^T + bias ), W is (N x K) row-major.
// Hin is LDS, K..Kpad-1 columns must be zero-padded. M,N multiples of 16.
// One 16x16 output tile per wave iteration; all 32 lanes active (EXEC=~0).
// Fragment layouts per CDNA5 ISA 7.12.2 (32-bit A 16x4 / B 4x16 / C 16x16).
// ---------------------------------------------------------------------------
__device__ __forceinline__ void mlp_layer(const float* __restrict__ Hin, int M, int K, int Kpad,
                                          const float* __restrict__ W, const float* __restrict__ bias,
                                          float* __restrict__ Hout, int N,
                                          int lane, int wave, int nwaves)
{
    const int mt = M >> 4, nt = N >> 4;
    const int T = mt * nt;
    const int r  = lane & 15;        // row/col within half-wave
    const int hi = lane >> 4;        // 0: K pair {0,1}, 1: K pair {2,3}
    for (int t = wave; t < T; t += nwaves) {
        const int m0 = (t % mt) << 4;
        const int n0 = (t / mt) << 4;
        v8f acc = {0.f, 0.f, 0.f, 0.f, 0.f, 0.f, 0.f, 0.f};
        const float* arow = Hin + (m0 + r) * Kpad + hi * 2;
        const float* brow = W + (size_t)(n0 + r) * K + hi * 2;
        for (int k = 0; k < Kpad; k += 4) {
            v2f a, b;
            a.x = arow[k];
            a.y = arow[k + 1];
            const int k0 = k + hi * 2;
            b.x = (k0     < K) ? brow[k]     : 0.f;
            b.y = (k0 + 1 < K) ? brow[k + 1] : 0.f;
            acc = __builtin_amdgcn_wmma_f32_16x16x4_f32(false, a, false, b,
                                                        (short)0, acc, false, false);
        }
        const int col = n0 + r;
        const float bv = bias[col];
        float* orow = Hout + (size_t)(m0 + hi * 8) * N + col;
#pragma unroll
        for (int rr = 0; rr < 8; ++rr) {
            float v = acc[rr] + bv;
            orow[(size_t)rr * N] = v > 0.f ? v : 0.f;   // ReLU
        }
    }
}

// ---------------------------------------------------------------------------
// FPS over 16384 points (coords = xyz[...,0:3]); one block per batch.
// dist[] lives in LDS (64KB); coords re-read from L2-resident global.
// Emits 32 centroid coords per batch (sample i = `far` entering step i).
// ---------------------------------------------------------------------------
__global__ void fps1_kernel(const float* __restrict__ xyz, float* __restrict__ new_xyz1)
{
    extern __shared__ float sm[];
    float* dist = sm;                  // NPTS
    float* redv = sm + NPTS;           // 256
    int*   redi = (int*)(redv + 256);  // 256
    const int b = blockIdx.x, tid = threadIdx.x;
    const float* base = xyz + (size_t)b * NPTS * 6;
    for (int p = tid; p < NPTS; p += 256) dist[p] = 1e10f;
    __syncthreads();
    int far = 0;
    for (int it = 0; it < 32; ++it) {
        const float cx = base[(size_t)far * 6 + 0];
        const float cy = base[(size_t)far * 6 + 1];
        const float cz = base[(size_t)far * 6 + 2];
        if (tid == 0) {
            new_xyz1[(b * 32 + it) * 3 + 0] = cx;
            new_xyz1[(b * 32 + it) * 3 + 1] = cy;
            new_xyz1[(b * 32 + it) * 3 + 2] = cz;
        }
        float bestv = -1.f; int besti = 0;
        for (int p = tid; p < NPTS; p += 256) {
            const float dx = base[(size_t)p * 6 + 0] - cx;
            const float dy = base[(size_t)p * 6 + 1] - cy;
            const float dz = base[(size_t)p * 6 + 2] - cz;
            float d = fminf(dist[p], dx * dx + dy * dy + dz * dz);
            dist[p] = d;
            if (d > bestv) { bestv = d; besti = p; }   // ascending p: first max wins
        }
        redv[tid] = bestv; redi[tid] = besti;
        __syncthreads();
        for (int off = 128; off > 0; off >>= 1) {
            if (tid < off) {
                const float ov = redv[tid + off]; const int oi = redi[tid + off];
                if (ov > redv[tid] || (ov == redv[tid] && oi < redi[tid])) {
                    redv[tid] = ov; redi[tid] = oi;
                }
            }
            __syncthreads();
        }
        far = redi[0];
        __syncthreads();
    }
}

// ---------------------------------------------------------------------------
// Ball query sa1: one wave per (b,s); first 32 in-radius indices in ascending
// index order via wave32 ballot prefix scan; pad with first found index.
// ---------------------------------------------------------------------------
__global__ void bq1_kernel(const float* __restrict__ xyz, const float* __restrict__ new_xyz1,
                           int* __restrict__ gi1)
{
    const int gwave = (blockIdx.x * blockDim.x + threadIdx.x) >> 5;
    const int lane = threadIdx.x & 31;
    if (gwave >= BATCH * 32) return;
    const int b = gwave >> 5;
    const float* base = xyz + (size_t)b * NPTS * 6;
    const float cx = new_xyz1[gwave * 3 + 0];
    const float cy = new_xyz1[gwave * 3 + 1];
    const float cz = new_xyz1[gwave * 3 + 2];
    int* out = gi1 + gwave * 32;
    const float r2 = 0.2f * 0.2f;
    int cnt = 0, first = -1;
    for (int bp = 0; bp < NPTS && cnt < 32; bp += 32) {
        const int p = bp + lane;
        const float dx = base[(size_t)p * 6 + 0] - cx;
        const float dy = base[(size_t)p * 6 + 1] - cy;
        const float dz = base[(size_t)p * 6 + 2] - cz;
        const bool inr = (dx * dx + dy * dy + dz * dz) <= r2;
        const unsigned int m = (unsigned int)__ballot(inr);
        if (first < 0 && m) first = bp + __ffs(m) - 1;
        if (inr) {
            const int pos = cnt + __popc(m & ((1u << lane) - 1u));
            if (pos < 32) out[pos] = p;
        }
        cnt += __popc(m);
    }
    for (int j = cnt + lane; j < 32; j += 32) out[j] = first;
}

// ---------------------------------------------------------------------------
// sa1 MLP: one block per (b,s). feat(32x8, K=6 padded) -> 64 -> 64 -> 128,
// then column-wise max over the 32 neighbors.
// ---------------------------------------------------------------------------
__global__ void sa1_mlp_kernel(const float* __restrict__ xyz, const float* __restrict__ new_xyz1,
                               const int* __restrict__ gi1,
                               const float* __restrict__ w1, const float* __restrict__ b1,
                               const float* __restrict__ w2, const float* __restrict__ b2,
                               const float* __restrict__ w3, const float* __restrict__ b3,
                               float* __restrict__ l1p)
{
    __shared__ float feat[32 * 8];
    __shared__ float h1[32 * 64];
    __shared__ float h2[32 * 64];
    __shared__ float h3[32 * 128];
    const int bs = blockIdx.x;            // b*32 + s
    const int b = bs >> 5;
    const int tid = threadIdx.x;
    const float* base = xyz + (size_t)b * NPTS * 6;
    const float* nx = new_xyz1 + bs * 3;
    const int* gi = gi1 + bs * 32;
    {
        const int row = tid >> 3, col = tid & 7;
        const int p = gi[row];
        float v = 0.f;
        if (col < 3)      v = base[(size_t)p * 6 + 3 + col];                 // normals first
        else if (col < 6) v = base[(size_t)p * 6 + (col - 3)] - nx[col - 3]; // then g_xyz
        feat[row * 8 + col] = v;
    }
    __syncthreads();
    const int lane = tid & 31, wave = tid >> 5;
    mlp_layer(feat, 32, 6, 8, w1, b1, h1, 64, lane, wave, 8);
    __syncthreads();
    mlp_layer(h1, 32, 64, 64, w2, b2, h2, 64, lane, wave, 8);
    __syncthreads();
    mlp_layer(h2, 32, 64, 64, w3, b3, h3, 128, lane, wave, 8);
    __syncthreads();
    if (tid < 128) {
        float m = h3[tid];
        for (int r = 1; r < 32; ++r) m = fmaxf(m, h3[r * 128 + tid]);
        l1p[(size_t)bs * 128 + tid] = m;
    }
}

// ---------------------------------------------------------------------------
// FPS over the 32 l1 points (one wave per batch, dist in registers).
// ---------------------------------------------------------------------------
__global__ void fps2_kernel(const float* __restrict__ l1x, float* __restrict__ new_xyz2)
{
    const int b = blockIdx.x, lane = threadIdx.x;   // blockDim = 32
    const float* base = l1x + b * 32 * 3;
    const float px = base[lane * 3 + 0];
    const float py = base[lane * 3 + 1];
    const float pz = base[lane * 3 + 2];
    float dist = 1e10f;
    int far = 0;
    for (int it = 0; it < 16; ++it) {
        const float cx = __shfl(px, far), cy = __shfl(py, far), cz = __shfl(pz, far);
        if (lane == 0) {
            new_xyz2[(b * 16 + it) * 3 + 0] = cx;
            new_xyz2[(b * 16 + it) * 3 + 1] = cy;
            new_xyz2[(b * 16 + it) * 3 + 2] = cz;
        }
        const float dx = px - cx, dy = py - cy, dz = pz - cz;
        dist = fminf(dist, dx * dx + dy * dy + dz * dz);
        float v = dist; int i = lane;
        for (int off = 16; off > 0; off >>= 1) {
            const float ov = __shfl_down(v, off); const int oi = __shfl_down(i, off);
            if (ov > v || (ov == v && oi < i)) { v = ov; i = oi; }
        }
        far = __shfl(i, 0);
    }
}

// ---------------------------------------------------------------------------
// Ball query sa2: n=32 exactly one ballot chunk; one wave per (b,s).
// ---------------------------------------------------------------------------
__global__ void bq2_kernel(const float* __restrict__ l1x, const float* __restrict__ new_xyz2,
                           int* __restrict__ gi2)
{
    const int gwave = (blockIdx.x * blockDim.x + threadIdx.x) >> 5;
    const int lane = threadIdx.x & 31;
    if (gwave >= BATCH * 16) return;
    const int b = gwave >> 4;
    const float* base = l1x + b * 32 * 3;
    const float cx = new_xyz2[gwave * 3 + 0];
    const float cy = new_xyz2[gwave * 3 + 1];
    const float cz = new_xyz2[gwave * 3 + 2];
    const float r2 = 0.4f * 0.4f;
    const float dx = base[lane * 3 + 0] - cx;
    const float dy = base[lane * 3 + 1] - cy;
    const float dz = base[lane * 3 + 2] - cz;
    const bool inr = (dx * dx + dy * dy + dz * dz) <= r2;
    const unsigned int m = (unsigned int)__ballot(inr);
    const int first = __ffs(m) - 1;          // m != 0 (center is a member point)
    const int cnt = __popc(m);
    int* out = gi2 + gwave * 32;
    if (inr) out[__popc(m & ((1u << lane) - 1u))] = lane;
    for (int j = cnt + lane; j < 32; j += 32) out[j] = first;
}

// ---------------------------------------------------------------------------
// sa2 MLP: one block per (b,s). feat(32x132, K=131) -> 128 -> 128 -> 256, max.
// ---------------------------------------------------------------------------
__global__ void sa2_mlp_kernel(const float* __restrict__ l1x, const float* __restrict__ l1p,
                               const float* __restrict__ new_xyz2, const int* __restrict__ gi2,
                               const float* __restrict__ w1, const float* __restrict__ b1,
                               const float* __restrict__ w2, const float* __restrict__ b2,
                               const float* __restrict__ w3, const float* __restrict__ b3,
                               float* __restrict__ l2p)
{
    extern __shared__ float sm[];
    float* feat = sm;                       // 32*132
    float* h1 = feat + 32 * 132;            // 32*128
    float* h2 = h1 + 32 * 128;              // 32*128
    float* h3 = h2 + 32 * 128;              // 32*256
    const int bs = blockIdx.x;              // b*16 + s
    const int b = bs >> 4;
    const int tid = threadIdx.x;
    const int* gi = gi2 + bs * 32;
    const float* nx = new_xyz2 + bs * 3;
    for (int e = tid; e < 32 * 132; e += 256) {
        const int row = e / 132, c = e % 132;
        const int p = gi[row];
        float v = 0.f;
        if (c < 128)      v = l1p[((size_t)b * 32 + p) * 128 + c];              // points first
        else if (c < 131) v = l1x[((size_t)b * 32 + p) * 3 + (c - 128)] - nx[c - 128];
        feat[e] = v;
    }
    __syncthreads();
    const int lane = tid & 31, wave = tid >> 5;
    mlp_layer(feat, 32, 131, 132, w1, b1, h1, 128, lane, wave, 8);
    __syncthreads();
    mlp_layer(h1, 32, 128, 128, w2, b2, h2, 128, lane, wave, 8);
    __syncthreads();
    mlp_layer(h2, 32, 128, 128, w3, b3, h3, 256, lane, wave, 8);
    __syncthreads();
    {
        float m = h3[tid];
        for (int r = 1; r < 32; ++r) m = fmaxf(m, h3[r * 256 + tid]);
        l2p[(size_t)bs * 256 + tid] = m;
    }
}

// ---------------------------------------------------------------------------
// sa3 group-all + FC: one block per batch. feat(16x260, K=259) -> 256 -> 512
// -> 128, max over 16, then 128->40 dense.
// ---------------------------------------------------------------------------
__global__ void sa3_fc_kernel(const float* __restrict__ l2x, const float* __restrict__ l2p,
                              const float* __restrict__ w1, const float* __restrict__ b1,
                              const float* __restrict__ w2, const float* __restrict__ b2,
                              const float* __restrict__ w3, const float* __restrict__ b3,
                              const float* __restrict__ fcw, const float* __restrict__ fcb,
                              float* __restrict__ outp)
{
    extern __shared__ float sm[];
    float* feat = sm;                       // 16*260
    float* h1 = feat + 16 * 260;            // 16*256
    float* h2 = h1 + 16 * 256;              // 16*512
    float* h3 = h2 + 16 * 512;              // 16*128
    float* l3 = h3 + 16 * 128;              // 128
    const int b = blockIdx.x;
    const int tid = threadIdx.x;
    for (int e = tid; e < 16 * 260; e += 256) {
        const int row = e / 260, c = e % 260;
        float v = 0.f;
        if (c < 3)        v = l2x[((size_t)b * 16 + row) * 3 + c];              // xyz first
        else if (c < 259) v = l2p[((size_t)b * 16 + row) * 256 + (c - 3)];
        feat[e] = v;
    }
    __syncthreads();
    const int lane = tid & 31, wave = tid >> 5;
    mlp_layer(feat, 16, 259, 260, w1, b1, h1, 256, lane, wave, 8);
    __syncthreads();
    mlp_layer(h1, 16, 256, 256, w2, b2, h2, 512, lane, wave, 8);
    __syncthreads();
    mlp_layer(h2, 16, 512, 512, w3, b3, h3, 128, lane, wave, 8);
    __syncthreads();
    if (tid < 128) {
        float m = h3[tid];
        for (int r = 1; r < 16; ++r) m = fmaxf(m, h3[r * 128 + tid]);
        l3[tid] = m;
    }
    __syncthreads();
    if (tid < 40) {
        float acc = fcb[tid];
        for (int c = 0; c < 128; ++c) acc += l3[c] * fcw[tid * 128 + c];
        outp[b * 40 + tid] = acc;
    }
}

// ---------------------------------------------------------------------------
extern "C" void kernel_launch(void* const* d_in, const int* in_sizes, int n_in,
                              void* d_out, int out_size, void* d_ws, size_t ws_size,
                              hipStream_t stream)
{
    (void)in_sizes; (void)n_in; (void)out_size; (void)ws_size;
    const float* xyz = (const float*)d_in[0];
    const float* w[9]; const float* bb[9];
    for (int i = 0; i < 9; ++i) { w[i] = (const float*)d_in[1 + 2 * i]; bb[i] = (const float*)d_in[2 + 2 * i]; }
    const float* fcw = (const float*)d_in[19];
    const float* fcb = (const float*)d_in[20];
    float* ws = (float*)d_ws;
    float* new_xyz1 = ws;                       // 64*32*3   = 6144 f
    int*   gi1      = (int*)(ws + 6144);        // 64*32*32  = 65536 i
    float* l1p      = ws + 71680;               // 64*32*128 = 262144 f
    float* new_xyz2 = ws + 333824;              // 64*16*3   = 3072 f
    int*   gi2      = (int*)(ws + 334592);      // 64*16*32  = 32768 i
    float* l2p      = ws + 367360;              // 64*16*256 = 262144 f
    float* out = (float*)d_out;

    fps1_kernel<<<BATCH, 256, (NPTS + 512) * 4, stream>>>(xyz, new_xyz1);
    bq1_kernel<<<(BATCH * 32 * 32) / 256, 256, 0, stream>>>(xyz, new_xyz1, gi1);
    sa1_mlp_kernel<<<BATCH * 32, 256, 0, stream>>>(xyz, new_xyz1, gi1,
                                                   w[0], bb[0], w[1], bb[1], w[2], bb[2], l1p);
    fps2_kernel<<<BATCH, 32, 0, stream>>>(new_xyz1, new_xyz2);
    bq2_kernel<<<(BATCH * 16 * 32) / 256, 256, 0, stream>>>(new_xyz1, new_xyz2, gi2);
    sa2_mlp_kernel<<<BATCH * 16, 256, (32 * 132 + 32 * 128 * 2 + 32 * 256) * 4, stream>>>(
        new_xyz1, l1p, new_xyz2, gi2, w[3], bb[3], w[4], bb[4], w[5], bb[5], l2p);
    sa3_fc_kernel<<<BATCH, 256, (16 * 260 + 16 * 256 + 16 * 512 + 16 * 128 + 128) * 4, stream>>>(
        new_xyz2, l2p, w[6], bb[6], w[7], bb[7], w[8], bb[8], fcw, fcb, out);
}